// Adam_80796924772652
// MI455X (gfx1250) — compile-verified
//
#include <hip/hip_runtime.h>
#include <stdint.h>

// Adam scalar recurrence: strictly sequential, latency-bound.
// One wave: lane 0 runs the recurrence in registers, staging each tile of
// (v, w) pairs into LDS (one ds_store_b64 per step); all 32 lanes then stream
// the tile to global memory with CDNA5 async LDS->global stores (ASYNCcnt),
// double-buffered so stores overlap the next tile's compute.
//
// Latency optimization: once beta1^t and beta2^t shrink below 2^-25 in fp32,
// (1 - beta^t) == 1.0f exactly, so the bias-correction divisions are divisions
// by exactly 1.0 (bitwise identity). We detect that tile boundary with a
// wave-uniform flag and switch to a fast loop that skips both divisions,
// producing bitwise-identical results with a much shorter dependency chain.

#define TILE 256  // steps per tile; 2 buffers * 256 * 8B = 4 KB LDS

__global__ __launch_bounds__(32)
void Adam_80796924772652_kernel(const float* __restrict__ beta1p,
                                const float* __restrict__ beta2p,
                                const float* __restrict__ lrp,
                                const int*   __restrict__ turnsp,
                                float*       __restrict__ out) {
  __shared__ float2 st[2][TILE];   // {v, w} pairs, 8B aligned for ds_store_b64

  const int   lane  = threadIdx.x;
  const float beta1 = beta1p[0];
  const float beta2 = beta2p[0];
  const float lr    = lrp[0];
  const int   turns = turnsp[0];

  float* gout = out;                  // v trajectory, length turns+1
  float* wout = out + (turns + 1);    // w trajectory, length turns+1

  if (lane == 0) {
    gout[0] = 0.0f;   // reference prepends initial (0, 0)
    wout[0] = 0.0f;
  }

  // Recurrence state lives in lane 0's VGPRs.
  float w = 0.0f, v = 0.0f, s = 0.0f;
  float p1 = 1.0f, p2 = 1.0f;               // beta1^t, beta2^t (incremental)
  const float om1 = 1.0f - beta1;
  const float om2 = 1.0f - beta2;

  int t    = 0;
  int buf  = 0;
  int fast = 0;   // wave-uniform: bias corrections have become exact no-ops
  while (t < turns) {
    const int n = (turns - t) < TILE ? (turns - t) : TILE;

    // This buffer was the async-store source two tiles ago. Each full tile
    // issues exactly 16 async stores; async stores complete in order, so
    // ASYNCcnt <= 16 guarantees this buffer's stores have drained.
    asm volatile("s_wait_asynccnt 16" ::: "memory");

    if (lane == 0) {
      float lw = w, lv = v, ls = s;
      if (!fast) {
        // Full Adam step with bias correction.
        float lp1 = p1, lp2 = p2;
        for (int j = 0; j < n; ++j) {
          const float g  = 2.0f * (lw - 3.0f);        // d/dw (w-3)^2
          ls = beta2 * ls + om2 * (g * g);            // second moment
          lv = beta1 * lv + om1 * g;                  // first moment
          lp1 *= beta1;
          lp2 *= beta2;
          const float vh = lv / (1.0f - lp1);         // bias correction
          const float sh = ls / (1.0f - lp2);
          lw = lw - lr * vh / (sqrtf(sh) + 1e-7f);
          st[buf][j] = make_float2(lv, lw);           // one ds_store_b64
        }
        p1 = lp1; p2 = lp2;
      } else {
        // (1-p1)==1 and (1-p2)==1 exactly: v/1.0==v, s/1.0==s bitwise.
        for (int j = 0; j < n; ++j) {
          const float g  = 2.0f * (lw - 3.0f);
          ls = beta2 * ls + om2 * (g * g);
          lv = beta1 * lv + om1 * g;
          lw = lw - lr * lv / (sqrtf(ls) + 1e-7f);
          st[buf][j] = make_float2(lv, lw);
        }
      }
      w = lw; v = lv; s = ls;
    }

    // Broadcast lane 0's phase predicate so the tile-loop branch is scalar.
    {
      const int f = (((1.0f - p1) == 1.0f) && ((1.0f - p2) == 1.0f)) ? 1 : 0;
      fast = __builtin_amdgcn_readfirstlane(f);
    }

    // LDS writes must be visible before the async DMA engine reads LDS
    // (DS ops and async ops use separate counters -> explicit dscnt wait).
    asm volatile("s_wait_dscnt 0x0" ::: "memory");

    // All 32 lanes: coalesced async LDS -> global stores (tracked by ASYNCcnt).
    // Per-lane LDS addresses de-interleave the {v,w} pairs for free.
    for (int j = lane; j < n; j += 32) {
      const uint64_t ga = (uint64_t)(uintptr_t)(gout + t + 1 + j);
      const uint32_t la = (uint32_t)(uintptr_t)(&st[buf][j].x);  // addr[31:0] = LDS byte offset
      asm volatile("global_store_async_from_lds_b32 %0, %1, off"
                   :: "v"(ga), "v"(la) : "memory");
      const uint64_t wa = (uint64_t)(uintptr_t)(wout + t + 1 + j);
      const uint32_t lb = (uint32_t)(uintptr_t)(&st[buf][j].y);
      asm volatile("global_store_async_from_lds_b32 %0, %1, off"
                   :: "v"(wa), "v"(lb) : "memory");
    }

    t   += n;
    buf ^= 1;
  }

  // Drain outstanding async stores (S_ENDPGM also implies wait-idle).
  asm volatile("s_wait_asynccnt 0x0" ::: "memory");
}

extern "C" void kernel_launch(void* const* d_in, const int* in_sizes, int n_in,
                              void* d_out, int out_size, void* d_ws, size_t ws_size,
                              hipStream_t stream) {
  (void)in_sizes; (void)n_in; (void)out_size; (void)d_ws; (void)ws_size;
  const float* beta1 = (const float*)d_in[0];
  const float* beta2 = (const float*)d_in[1];
  const float* lr    = (const float*)d_in[2];
  const int*   turns = (const int*)d_in[3];
  float*       out   = (float*)d_out;

  // Single wave: the computation is one serial dependency chain.
  Adam_80796924772652_kernel<<<1, 32, 0, stream>>>(beta1, beta2, lr, turns, out);
}